// HHGLCM_31662498906541
// MI455X (gfx1250) — compile-verified
//
#include <hip/hip_runtime.h>
#include <math.h>

#define WAY 64
#define NQ 2048
#define CH 640
#define PP 5
#define EPSI 0.05f
#define SINK_ITERS 100
#define TEMP 12.5f
#define NPRIME 320   // WAY*PP
#define MPRIME 10240 // NQ*PP

typedef __attribute__((ext_vector_type(2))) float v2f;
typedef __attribute__((ext_vector_type(8))) float v8f;

// ---------------- Kernel 1: region pooling -> features [N][P][C] ----------
__global__ void pool_feats(const float* __restrict__ x, float* __restrict__ f, int N) {
  int t = blockIdx.x * blockDim.x + threadIdx.x;
  if (t >= N * CH) return;
  int m = t / CH, c = t % CH;
  const float* p = x + (size_t)t * 25;
  float v[25];
#pragma unroll
  for (int i = 0; i < 25; i++) v[i] = p[i];
  float lt = 0.f, rt = 0.f, mid = 0.f, lb = 0.f, rb = 0.f;
#pragma unroll
  for (int h = 0; h < 3; h++)
#pragma unroll
    for (int w = 0; w < 3; w++) {
      lt += v[h * 5 + w];
      rt += v[(h + 2) * 5 + w];
      lb += v[h * 5 + (w + 2)];
      rb += v[(h + 2) * 5 + (w + 2)];
    }
#pragma unroll
  for (int h = 1; h < 5; h++)
#pragma unroll
    for (int w = 1; w < 5; w++) mid += v[h * 5 + w];
  float* o = f + ((size_t)m * PP) * CH + c;
  o[0 * CH] = lt * (1.f / 9.f);
  o[1 * CH] = rt * (1.f / 9.f);
  o[2 * CH] = mid * (1.f / 16.f);
  o[3 * CH] = lb * (1.f / 9.f);
  o[4 * CH] = rb * (1.f / 9.f);
}

// ------- Kernel 2: channel-center + L2 normalize one (n,p) row of C=640 ----
// transposeOut: write out[c][row] with row-stride NPRIME (for B^T of the GEMM)
__global__ void center_norm(const float* __restrict__ in, float* __restrict__ out,
                            int transposeOut) {
  __shared__ float red[128];
  int row = blockIdx.x;
  int tid = threadIdx.x;
  const float* base = in + (size_t)row * CH;
  float v[5];
  float s = 0.f;
#pragma unroll
  for (int k = 0; k < 5; k++) { v[k] = base[tid + k * 128]; s += v[k]; }
  red[tid] = s;
  __syncthreads();
  for (int off = 64; off > 0; off >>= 1) {
    if (tid < off) red[tid] += red[tid + off];
    __syncthreads();
  }
  float mean = red[0] * (1.f / CH);
  __syncthreads();
  float ss = 0.f;
#pragma unroll
  for (int k = 0; k < 5; k++) { v[k] -= mean; ss += v[k] * v[k]; }
  red[tid] = ss;
  __syncthreads();
  for (int off = 64; off > 0; off >>= 1) {
    if (tid < off) red[tid] += red[tid + off];
    __syncthreads();
  }
  float sc = 1.f / fmaxf(sqrtf(red[0]), 1e-8f);
#pragma unroll
  for (int k = 0; k < 5; k++) {
    int c = tid + k * 128;
    float val = v[k] * sc;
    if (transposeOut)
      out[(size_t)c * NPRIME + row] = val;
    else
      out[(size_t)row * CH + c] = val;
  }
}

// ------- Kernel 3: weights  w[m][n][p] = relu(dot)+0.00101, row-norm to P ---
// one wave per (m,n) pair; lanes stride the C=640 dot product
__global__ void weights_kernel(const float* __restrict__ qf, const float* __restrict__ pf,
                               float* __restrict__ wb) {
  int gwave = (blockIdx.x * blockDim.x + threadIdx.x) >> 5;
  int lane = threadIdx.x & 31;
  int m = gwave / WAY;
  int n = gwave % WAY;
  float d[5];
#pragma unroll
  for (int p = 0; p < 5; p++) {
    const float* qr = qf + ((size_t)m * PP + p) * CH;
    const float* pr = pf + ((size_t)n * PP + p) * CH;
    float acc = 0.f;
    for (int c = lane; c < CH; c += 32) acc += qr[c] * pr[c];
#pragma unroll
    for (int off = 16; off > 0; off >>= 1) acc += __shfl_xor(acc, off, 32);
    d[p] = fmaxf(acc, 0.f) + 0.00101f;  // relu(comb)+0.001 then relu(+)+1e-5
  }
  float s = d[0] + d[1] + d[2] + d[3] + d[4];
  float sc = (float)PP / s;
  if (lane == 0) {
#pragma unroll
    for (int p = 0; p < 5; p++) wb[(size_t)gwave * PP + p] = d[p] * sc;
  }
}

// ------- Kernel 4: sim GEMM via V_WMMA_F32_16X16X4_F32 ---------------------
// C[10240][320] = A[10240][640] * BT[640][320].
// Each wave: 32x64 output tile = 2 (M) x 4 (N) accumulators.
// 8 WMMAs per K-step against 2 b64 + 8 b32 loads (A reused 4x, B reused 2x).
// 320 M-tiles * 5 N-supertiles = 1600 waves = 200 blocks x 256 (exact; no
// divergence -> EXEC all ones as WMMA requires).
__global__ void sim_wmma(const float* __restrict__ A, const float* __restrict__ BT,
                         float* __restrict__ Cout) {
  int waveId = threadIdx.x >> 5;
  int tile = blockIdx.x * 8 + waveId;        // [0, 1600)
  int tm = tile % (MPRIME / 32);             // 320 tiles along M'
  int tn = tile / (MPRIME / 32);             // 5 supertiles along N'
  int lane = threadIdx.x & 31;
  int half = lane >> 4;                      // 0: k0,k0+1   1: k0+2,k0+3
  int l16 = lane & 15;

  const float* arow[2];
#pragma unroll
  for (int mi = 0; mi < 2; mi++)
    arow[mi] = A + ((size_t)tm * 32 + mi * 16 + l16) * CH + 2 * half;
  const float* bcol[4];
#pragma unroll
  for (int nf = 0; nf < 4; nf++)
    bcol[nf] = BT + (size_t)tn * 64 + nf * 16 + l16 + (size_t)(2 * half) * NPRIME;

  v8f acc[2][4];
#pragma unroll
  for (int mi = 0; mi < 2; mi++)
#pragma unroll
    for (int nf = 0; nf < 4; nf++) acc[mi][nf] = (v8f){0.f, 0.f, 0.f, 0.f, 0.f, 0.f, 0.f, 0.f};

  for (int k0 = 0; k0 < CH; k0 += 4) {
    // prefetch one K-block ahead (global_prefetch_b8; speculative, safe at end)
    __builtin_prefetch(arow[0] + k0 + 64, 0, 1);
    __builtin_prefetch(bcol[0] + (size_t)(k0 + 64) * NPRIME, 0, 1);

    v2f a[2], b[4];
#pragma unroll
    for (int mi = 0; mi < 2; mi++) {
      a[mi][0] = arow[mi][k0];
      a[mi][1] = arow[mi][k0 + 1];
    }
#pragma unroll
    for (int nf = 0; nf < 4; nf++) {
      b[nf][0] = bcol[nf][(size_t)k0 * NPRIME];
      b[nf][1] = bcol[nf][(size_t)(k0 + 1) * NPRIME];
    }
#pragma unroll
    for (int mi = 0; mi < 2; mi++)
#pragma unroll
      for (int nf = 0; nf < 4; nf++)
        acc[mi][nf] = __builtin_amdgcn_wmma_f32_16x16x4_f32(
            false, a[mi], false, b[nf], (short)0, acc[mi][nf], false, false);
  }

#pragma unroll
  for (int mi = 0; mi < 2; mi++)
#pragma unroll
    for (int nf = 0; nf < 4; nf++)
#pragma unroll
      for (int v = 0; v < 8; v++) {
        Cout[(size_t)(tm * 32 + mi * 16 + v + 8 * half) * NPRIME + tn * 64 + nf * 16 +
             l16] = acc[mi][nf][v];
      }
}

// ------- Kernel 5: log-domain Sinkhorn (a == b) + fused logits -------------
__global__ void sinkhorn_logits(const float* __restrict__ sim, const float* __restrict__ wb,
                                float* __restrict__ out) {
  int t = blockIdx.x * blockDim.x + threadIdx.x;  // m*64 + w, t < 131072
  int m = t >> 6;
  int w = t & 63;
  const float inv_eps = 1.f / EPSI;

  float cost[5][5];
#pragma unroll
  for (int i = 0; i < 5; i++)
#pragma unroll
    for (int j = 0; j < 5; j++)
      cost[i][j] = 1.f - sim[(size_t)(m * 5 + i) * NPRIME + w * 5 + j];

  float lw[5];
#pragma unroll
  for (int p = 0; p < 5; p++) lw[p] = __logf(wb[(size_t)t * PP + p]);

  float f[5] = {0.f, 0.f, 0.f, 0.f, 0.f};
  float g[5] = {0.f, 0.f, 0.f, 0.f, 0.f};

  for (int it = 0; it < SINK_ITERS; it++) {
#pragma unroll
    for (int i = 0; i < 5; i++) {
      float arg[5], mx = -3.402823466e38f;
#pragma unroll
      for (int j = 0; j < 5; j++) {
        arg[j] = (g[j] - cost[i][j]) * inv_eps;
        mx = fmaxf(mx, arg[j]);
      }
      float s = 0.f;
#pragma unroll
      for (int j = 0; j < 5; j++) s += __expf(arg[j] - mx);
      f[i] = EPSI * (lw[i] - (mx + __logf(s)));
    }
#pragma unroll
    for (int j = 0; j < 5; j++) {
      float arg[5], mx = -3.402823466e38f;
#pragma unroll
      for (int i = 0; i < 5; i++) {
        arg[i] = (f[i] - cost[i][j]) * inv_eps;
        mx = fmaxf(mx, arg[i]);
      }
      float s = 0.f;
#pragma unroll
      for (int i = 0; i < 5; i++) s += __expf(arg[i] - mx);
      g[j] = EPSI * (lw[j] - (mx + __logf(s)));
    }
  }

  float acc = 0.f;
#pragma unroll
  for (int i = 0; i < 5; i++)
#pragma unroll
    for (int j = 0; j < 5; j++) {
      float flow = __expf((f[i] + g[j] - cost[i][j]) * inv_eps);
      acc += (1.f - cost[i][j]) * flow;
    }
  out[t] = acc * (TEMP / (float)PP);
}

extern "C" void kernel_launch(void* const* d_in, const int* in_sizes, int n_in,
                              void* d_out, int out_size, void* d_ws, size_t ws_size,
                              hipStream_t stream) {
  const float* proto = (const float*)d_in[0];  // (1,64,640,5,5)
  const float* query = (const float*)d_in[1];  // (2048,640,5,5)
  float* out = (float*)d_out;                  // (2048,64)
  float* ws = (float*)d_ws;

  float* qf = ws;                              // 2048*5*640
  float* pf = qf + (size_t)NQ * PP * CH;       // 64*5*640
  float* qn = pf + (size_t)WAY * PP * CH;      // 2048*5*640
  float* pnT = qn + (size_t)NQ * PP * CH;      // 640*320
  float* wb = pnT + (size_t)CH * NPRIME;       // 2048*64*5
  float* sm = wb + (size_t)NQ * WAY * PP;      // 10240*320

  pool_feats<<<(NQ * CH + 255) / 256, 256, 0, stream>>>(query, qf, NQ);
  pool_feats<<<(WAY * CH + 255) / 256, 256, 0, stream>>>(proto, pf, WAY);
  center_norm<<<NQ * PP, 128, 0, stream>>>(qf, qn, 0);
  center_norm<<<WAY * PP, 128, 0, stream>>>(pf, pnT, 1);
  weights_kernel<<<(NQ * WAY * 32) / 256, 256, 0, stream>>>(qf, pf, wb);
  sim_wmma<<<(MPRIME / 32) * (NPRIME / 64) / 8, 256, 0, stream>>>(qn, pnT, sm);
  sinkhorn_logits<<<(NQ * WAY) / 256, 256, 0, stream>>>(sm, wb, out);
}